// einnet_61220463837619
// MI455X (gfx1250) — compile-verified
//
#include <hip/hip_runtime.h>

// Batched (Z=32768) fused 32x32 matmul chain: y2 = (x1 @ x0) @ x2 per z,
// layout (i*32+j)*Z + z (z innermost). Memory-bound: 537 MB @ 23.3 TB/s ~= 23us.
// FP32 WMMA (v_wmma_f32_16x16x4_f32) keeps the 4.3 GFLOP well under that floor.
// x0/x2 are staged TRANSPOSED in LDS so every WMMA fragment (A and B) is a
// single 8B-aligned ds_load_b64 into a contiguous VGPR pair (no marshalling).

typedef __attribute__((ext_vector_type(2))) float v2f;
typedef __attribute__((ext_vector_type(4))) float v4f;
typedef __attribute__((ext_vector_type(8))) float v8f;

#define ZDIM    32768
#define ZT      4        // z-values per block (one per wave)
#define THREADS 128      // 4 wave32
#define RSTRIDE 36       // padded row stride: 36 = 4*(odd) mod 64 -> b64 frag reads conflict-free
#define SLICE   (32 * RSTRIDE)      // one z-slice of a 32x32 matrix in LDS
#define YSLICE  (16 * RSTRIDE)      // per-wave 16x32 y1 row-block staging
// LDS floats: 3 inputs * ZT slices + ZT y1 blocks = 3*4*1152 + 4*576 = 16128 -> 64512 B
#define SMEM_FLOATS (3 * ZT * SLICE + ZT * YSLICE)

// A fragment (16x4, MxK) for V_WMMA_F32_16X16X4_F32 from a row-major padded slice.
// Lanes 0-15: M=lane, K={k,k+1}; lanes 16-31: M=lane-16, K={k+2,k+3}. K pairs are
// row-adjacent -> single 8B-aligned v2f load.
__device__ __forceinline__ v2f load_a_frag(const float* slice, int mt, int k, int lane) {
    int m  = mt + (lane & 15);
    int kk = k + ((lane >> 4) << 1);
    return *(const v2f*)(slice + m * RSTRIDE + kk);
}

// B fragment (4x16, KxN) from a TRANSPOSED ([n][k]) padded slice.
// Lanes 0-15: K={k,k+1}, N=lane; lanes 16-31: K={k+2,k+3}, N=lane-16.
// K pairs are row-adjacent in the transposed slice -> single v2f load.
__device__ __forceinline__ v2f load_bT_frag(const float* sliceT, int k, int nt, int lane) {
    int n  = nt + (lane & 15);
    int kk = k + ((lane >> 4) << 1);
    return *(const v2f*)(sliceT + n * RSTRIDE + kk);
}

__global__ void __launch_bounds__(THREADS) einnet_batched_wmma(
    const float* __restrict__ x0,   // (b,c,Z)
    const float* __restrict__ x1,   // (a,b,Z)
    const float* __restrict__ x2,   // (c,d,Z)
    float* __restrict__ out) {      // (a,d,Z)
    extern __shared__ float smem[];
    float* s_x0 = smem;                     // ZT slices [c][b] (transposed)
    float* s_x1 = smem + ZT * SLICE;        // ZT slices [a][b] (row-major)
    float* s_x2 = smem + 2 * ZT * SLICE;    // ZT slices [d][c] (transposed)
    float* s_y1 = smem + 3 * ZT * SLICE;    // ZT blocks [16][RSTRIDE] (row-major)

    const int tid = threadIdx.x;
    const int z0  = blockIdx.x * ZT;

    // ---- Phase 1: coalesced global -> LDS stage (b128 loads, 4 z per lane) ----
    #pragma unroll
    for (int it = 0; it < 1024 / THREADS; ++it) {
        int row = tid + it * THREADS;           // row = i*32 + j
        long g  = (long)row * ZDIM + z0;
        int i = row >> 5, j = row & 31;
        int e  = i * RSTRIDE + j;               // row-major slot (x1)
        int et = j * RSTRIDE + i;               // transposed slot (x0, x2)
        v4f v0 = *(const v4f*)(x0 + g);
        v4f v1 = *(const v4f*)(x1 + g);
        v4f v2 = *(const v4f*)(x2 + g);
        #pragma unroll
        for (int zz = 0; zz < ZT; ++zz) {
            s_x0[zz * SLICE + et] = v0[zz];
            s_x1[zz * SLICE + e]  = v1[zz];
            s_x2[zz * SLICE + et] = v2[zz];
        }
    }
    __syncthreads();

    // ---- Phase 2: per-wave fused matmul chain for its z slice ----
    const int lane = tid & 31;
    const int zl   = tid >> 5;                  // wave id == z slice
    const float* m_x0 = s_x0 + zl * SLICE;
    const float* m_x1 = s_x1 + zl * SLICE;
    const float* m_x2 = s_x2 + zl * SLICE;
    float*       yb   = s_y1 + zl * YSLICE;
    const int rrow = (lane >> 4) << 3;          // C/D: lanes16-31 hold M=r+8
    const int ncol = lane & 15;

    v8f y2t[2][2];                              // y2 tiles [mt][nt] kept in registers
    #pragma unroll
    for (int mi = 0; mi < 2; ++mi) {
        const int mt = mi * 16;
        // matmul1: y1[mt..mt+15][0..31] = x1[mt..][:] @ x0
        v8f acc0 = {}, acc1 = {};
        #pragma unroll
        for (int k = 0; k < 32; k += 4) {
            v2f a  = load_a_frag(m_x1, mt, k, lane);
            v2f b0 = load_bT_frag(m_x0, k, 0, lane);
            v2f b1 = load_bT_frag(m_x0, k, 16, lane);
            acc0 = __builtin_amdgcn_wmma_f32_16x16x4_f32(false, a, false, b0,
                                                         (short)0, acc0, false, false);
            acc1 = __builtin_amdgcn_wmma_f32_16x16x4_f32(false, a, false, b1,
                                                         (short)0, acc1, false, false);
        }
        // Stage y1 row-block (D layout -> row-major) in wave-private LDS.
        #pragma unroll
        for (int r = 0; r < 8; ++r) {
            yb[(r + rrow) * RSTRIDE + ncol]      = acc0[r];
            yb[(r + rrow) * RSTRIDE + 16 + ncol] = acc1[r];
        }
        // Same-wave LDS RAW: DS ops are in-order per wave; wait before VALU use.
        asm volatile("s_wait_dscnt 0" ::: "memory");
        // matmul2: y2[mt..mt+15][:] = y1row @ x2
        v8f d0 = {}, d1 = {};
        #pragma unroll
        for (int k = 0; k < 32; k += 4) {
            v2f a  = load_a_frag(yb, 0, k, lane);
            v2f b0 = load_bT_frag(m_x2, k, 0, lane);
            v2f b1 = load_bT_frag(m_x2, k, 16, lane);
            d0 = __builtin_amdgcn_wmma_f32_16x16x4_f32(false, a, false, b0,
                                                       (short)0, d0, false, false);
            d1 = __builtin_amdgcn_wmma_f32_16x16x4_f32(false, a, false, b1,
                                                       (short)0, d1, false, false);
        }
        y2t[mi][0] = d0;
        y2t[mi][1] = d1;
    }
    __syncthreads();   // everyone done reading input tiles

    // ---- Phase 3: deposit y2 into LDS as [row][ZT] (reuse s_x0 region) ----
    #pragma unroll
    for (int mi = 0; mi < 2; ++mi) {
        #pragma unroll
        for (int ni = 0; ni < 2; ++ni) {
            #pragma unroll
            for (int r = 0; r < 8; ++r) {
                int ar = mi * 16 + r + rrow;
                int dc = ni * 16 + ncol;
                smem[(ar * 32 + dc) * ZT + zl] = y2t[mi][ni][r];
            }
        }
    }
    __syncthreads();

    // ---- Phase 4: coalesced LDS -> global store (b128, 4 z per lane) ----
    #pragma unroll
    for (int it = 0; it < 1024 / THREADS; ++it) {
        int row = tid + it * THREADS;
        long g  = (long)row * ZDIM + z0;
        *(v4f*)(out + g) = ((const v4f*)smem)[row];
    }
}

extern "C" void kernel_launch(void* const* d_in, const int* in_sizes, int n_in,
                              void* d_out, int out_size, void* d_ws, size_t ws_size,
                              hipStream_t stream) {
    (void)in_sizes; (void)n_in; (void)d_ws; (void)ws_size; (void)out_size;
    const float* x0 = (const float*)d_in[0];
    const float* x1 = (const float*)d_in[1];
    const float* x2 = (const float*)d_in[2];
    float* out = (float*)d_out;
    einnet_batched_wmma<<<ZDIM / ZT, THREADS, SMEM_FLOATS * sizeof(float), stream>>>(
        x0, x1, x2, out);
}